// LinearAttnDecodeStep_62680752718085
// MI455X (gfx1250) — compile-verified
//
#include <hip/hip_runtime.h>
#include <math.h>

#define HDIM 2048
#define NV 32
#define NK 16
#define DK 128
#define DV 128
#define KEY_DIM 2048
#define VALUE_DIM 4096
#define CONV_DIM 8192
#define ROWS_TOTAL (CONV_DIM + VALUE_DIM + NV + NV) /* 12352 */

typedef float v2f __attribute__((ext_vector_type(2)));
typedef float v8f __attribute__((ext_vector_type(8)));

__device__ __forceinline__ float siluf(float x) { return x / (1.0f + expf(-x)); }

// ---------------------------------------------------------------------------
// K1: fused GEMV for qkv/z/b/a + depthwise-conv epilogue + new_conv_state.
// One row per wave (wave32), x staged in LDS, float4 streaming loads.
// ---------------------------------------------------------------------------
__global__ __launch_bounds__(256) void k1_gemv_conv(
    const float* __restrict__ x, const float* __restrict__ conv_state,
    const float* __restrict__ conv_w, const float* __restrict__ qkv_w,
    const float* __restrict__ z_w, const float* __restrict__ b_w,
    const float* __restrict__ a_w, float* __restrict__ out_conv_state,
    float* __restrict__ ws_conv_out, float* __restrict__ ws_z,
    float* __restrict__ ws_b, float* __restrict__ ws_a) {
  __shared__ __align__(16) float xs[HDIM];
  int tid = threadIdx.x;
#pragma unroll
  for (int i = 0; i < HDIM / 256; ++i) xs[tid + i * 256] = x[tid + i * 256];
  __syncthreads();
  int wave = tid >> 5, lane = tid & 31;
  int row = blockIdx.x * 8 + wave;
  if (row >= ROWS_TOTAL) return;
  const float* wrow;
  if (row < CONV_DIM)                    wrow = qkv_w + (size_t)row * HDIM;
  else if (row < CONV_DIM + VALUE_DIM)   wrow = z_w + (size_t)(row - CONV_DIM) * HDIM;
  else if (row < CONV_DIM + VALUE_DIM + NV) wrow = b_w + (size_t)(row - CONV_DIM - VALUE_DIM) * HDIM;
  else                                   wrow = a_w + (size_t)(row - CONV_DIM - VALUE_DIM - NV) * HDIM;
  const float4* w4 = (const float4*)wrow;
  const float4* x4 = (const float4*)xs;
  float sum = 0.0f;
#pragma unroll
  for (int j = 0; j < HDIM / 128; ++j) {
    float4 wv = w4[lane + j * 32];
    float4 xv = x4[lane + j * 32];
    sum += wv.x * xv.x + wv.y * xv.y + wv.z * xv.z + wv.w * xv.w;
  }
#pragma unroll
  for (int off = 16; off > 0; off >>= 1) sum += __shfl_xor(sum, off, 32);
  if (lane == 0) {
    if (row < CONV_DIM) {
      float cs1 = conv_state[row * 4 + 1];
      float cs2 = conv_state[row * 4 + 2];
      float cs3 = conv_state[row * 4 + 3];
      float co = cs1 * conv_w[row * 4 + 0] + cs2 * conv_w[row * 4 + 1] +
                 cs3 * conv_w[row * 4 + 2] + sum * conv_w[row * 4 + 3];
      ws_conv_out[row] = siluf(co);
      ((float4*)out_conv_state)[row] = make_float4(cs1, cs2, cs3, sum);
    } else if (row < CONV_DIM + VALUE_DIM) {
      ws_z[row - CONV_DIM] = sum;
    } else if (row < CONV_DIM + VALUE_DIM + NV) {
      ws_b[row - CONV_DIM - VALUE_DIM] = sum;
    } else {
      ws_a[row - CONV_DIM - VALUE_DIM - NV] = sum;
    }
  }
}

// ---------------------------------------------------------------------------
// K2: l2norm of q/k (one wave per NK head) + beta / exp(g) scalars.
// ---------------------------------------------------------------------------
__global__ __launch_bounds__(512) void k2_norm_gates(
    const float* __restrict__ ws_conv_out, const float* __restrict__ ws_b,
    const float* __restrict__ ws_a, const float* __restrict__ dt_bias,
    const float* __restrict__ A_log, float* __restrict__ ws_qn,
    float* __restrict__ ws_kn, float* __restrict__ ws_beta,
    float* __restrict__ ws_gexp) {
  int tid = threadIdx.x, wave = tid >> 5, lane = tid & 31;  // 16 waves = NK
  const float4* q4 = (const float4*)(ws_conv_out + wave * DK);
  const float4* k4 = (const float4*)(ws_conv_out + KEY_DIM + wave * DK);
  float4 qv = q4[lane];
  float4 kv = k4[lane];
  float sq = qv.x * qv.x + qv.y * qv.y + qv.z * qv.z + qv.w * qv.w;
  float sk = kv.x * kv.x + kv.y * kv.y + kv.z * kv.z + kv.w * kv.w;
#pragma unroll
  for (int off = 16; off > 0; off >>= 1) {
    sq += __shfl_xor(sq, off, 32);
    sk += __shfl_xor(sk, off, 32);
  }
  float qs = rsqrtf(sq + 1e-6f) * 0.08838834764831845f;  // 1/sqrt(128)
  float ks = rsqrtf(sk + 1e-6f);
  float4 qo = make_float4(qv.x * qs, qv.y * qs, qv.z * qs, qv.w * qs);
  float4 ko = make_float4(kv.x * ks, kv.y * ks, kv.z * ks, kv.w * ks);
  ((float4*)(ws_qn + wave * DK))[lane] = qo;
  ((float4*)(ws_kn + wave * DK))[lane] = ko;
  if (tid < NV) {
    float aa = ws_a[tid] + dt_bias[tid];
    float sp = (aa > 30.0f) ? aa : log1pf(expf(aa));
    float g = -expf(A_log[tid]) * sp;
    ws_beta[tid] = 1.0f / (1.0f + expf(-ws_b[tid]));
    ws_gexp[tid] = expf(g);
  }
}

// ---------------------------------------------------------------------------
// K3: per-head recurrent update via V_WMMA_F32_16X16X4_F32.
//   y0 = rec^T k, y1 = rec^T q in one WMMA sweep (A row0=k, row1=q);
//   delta = (v - ge*y0)*beta; core = ge*y1 + (q.k)*delta;
//   rec_out = ge*rec + k (x) delta;  xn = rmsnorm(core)*norm_w*silu(z).
// One block per head (32 blocks, 8 waves); wave w owns N-tile [16w,16w+16).
// ---------------------------------------------------------------------------
__global__ __launch_bounds__(256) void k3_rec_wmma(
    const float* __restrict__ rec_in, const float* __restrict__ ws_conv_out,
    const float* __restrict__ ws_z, const float* __restrict__ ws_qn,
    const float* __restrict__ ws_kn, const float* __restrict__ ws_beta,
    const float* __restrict__ ws_gexp, const float* __restrict__ norm_w,
    float* __restrict__ rec_out, float* __restrict__ ws_xn) {
  __shared__ __align__(16) float y0S[DV];
  __shared__ __align__(16) float y1S[DV];
  __shared__ __align__(16) float coreS[DV];
  __shared__ __align__(16) float kS[DK];
  __shared__ __align__(16) float qS[DK];
  __shared__ float sS[2];  // [0]=q.k  [1]=inv rms
  int h = blockIdx.x;
  int tid = threadIdx.x, wave = tid >> 5, lane = tid & 31;
  float ge = ws_gexp[h];
  if (tid < DK) {
    kS[tid] = ws_kn[(h >> 1) * DK + tid];
    qS[tid] = ws_qn[(h >> 1) * DK + tid];
  }
  __syncthreads();
  if (wave == 0) {  // s = q.k (wave-uniform branch)
    float p = kS[lane] * qS[lane] + kS[lane + 32] * qS[lane + 32] +
              kS[lane + 64] * qS[lane + 64] + kS[lane + 96] * qS[lane + 96];
#pragma unroll
    for (int off = 16; off > 0; off >>= 1) p += __shfl_xor(p, off, 32);
    if (lane == 0) sS[0] = p;
  }
  // --- WMMA sweep: D(16x16) += A(16x4) x B(4x16), K chunked by 4 ---
  const float* rg = rec_in + (size_t)h * DK * DV;
  int m = lane & 15;
  int khalf = (lane >> 4) << 1;  // lanes 0-15: K offset 0; lanes 16-31: +2
  int ncol = wave * 16 + m;
  const float* aPtr = (m == 0) ? kS : qS;  // row0 = k, row1 = q
  float amask = (m < 2) ? 1.0f : 0.0f;
  v8f acc = {0.0f, 0.0f, 0.0f, 0.0f, 0.0f, 0.0f, 0.0f, 0.0f};
#pragma unroll 4
  for (int kc = 0; kc < DK; kc += 4) {
    int kk = kc + khalf;
    v2f a;
    a.x = aPtr[kk] * amask;
    a.y = aPtr[kk + 1] * amask;
    v2f b;
    b.x = rg[kk * DV + ncol];
    b.y = rg[(kk + 1) * DV + ncol];
    acc = __builtin_amdgcn_wmma_f32_16x16x4_f32(false, a, false, b, (short)0,
                                                acc, false, false);
  }
  if (lane < 16) {  // D row M=0 -> acc[0], row M=1 -> acc[1] (lanes 0-15)
    y0S[ncol] = acc[0] * ge;
    y1S[ncol] = acc[1] * ge;
  }
  __syncthreads();
  float beta = ws_beta[h];
  if (tid < DV) {
    float vv = ws_conv_out[2 * KEY_DIM + h * DV + tid];
    float d = (vv - y0S[tid]) * beta;
    float core = y1S[tid] + sS[0] * d;
    y0S[tid] = d;  // reuse as delta
    coreS[tid] = core;
  }
  __syncthreads();
  if (wave == 0) {
    float c0 = coreS[lane], c1 = coreS[lane + 32];
    float c2 = coreS[lane + 64], c3 = coreS[lane + 96];
    float ss = c0 * c0 + c1 * c1 + c2 * c2 + c3 * c3;
#pragma unroll
    for (int off = 16; off > 0; off >>= 1) ss += __shfl_xor(ss, off, 32);
    if (lane == 0) sS[1] = rsqrtf(ss * (1.0f / DV) + 1e-6f);
  }
  __syncthreads();
  // rec_out = ge*rec + k (x) delta   (rec re-read from L2; coalesced in v)
  int v = tid & 127;
  int kbase = (tid >> 7) * 64;
  float d = y0S[v];
  float* ro = rec_out + (size_t)h * DK * DV;
#pragma unroll 4
  for (int k = kbase; k < kbase + 64; ++k)
    ro[k * DV + v] = ge * rg[k * DV + v] + kS[k] * d;
  if (tid < DV) {
    float zz = ws_z[h * DV + tid];
    ws_xn[h * DV + tid] = coreS[tid] * sS[1] * norm_w[tid] * siluf(zz);
  }
}

// ---------------------------------------------------------------------------
// K4: hidden_out = out_proj_w (2048x4096) @ xn
// ---------------------------------------------------------------------------
__global__ __launch_bounds__(256) void k4_outproj(
    const float* __restrict__ out_w, const float* __restrict__ ws_xn,
    float* __restrict__ hidden_out) {
  __shared__ __align__(16) float xs[VALUE_DIM];
  int tid = threadIdx.x;
#pragma unroll
  for (int i = 0; i < VALUE_DIM / 256; ++i)
    xs[tid + i * 256] = ws_xn[tid + i * 256];
  __syncthreads();
  int wave = tid >> 5, lane = tid & 31;
  int row = blockIdx.x * 8 + wave;
  if (row >= HDIM) return;
  const float4* w4 = (const float4*)(out_w + (size_t)row * VALUE_DIM);
  const float4* x4 = (const float4*)xs;
  float sum = 0.0f;
#pragma unroll
  for (int j = 0; j < VALUE_DIM / 128; ++j) {
    float4 wv = w4[lane + j * 32];
    float4 xv = x4[lane + j * 32];
    sum += wv.x * xv.x + wv.y * xv.y + wv.z * xv.z + wv.w * xv.w;
  }
#pragma unroll
  for (int off = 16; off > 0; off >>= 1) sum += __shfl_xor(sum, off, 32);
  if (lane == 0) hidden_out[row] = sum;
}

extern "C" void kernel_launch(void* const* d_in, const int* in_sizes, int n_in,
                              void* d_out, int out_size, void* d_ws,
                              size_t ws_size, hipStream_t stream) {
  const float* hidden_in  = (const float*)d_in[0];
  const float* conv_state = (const float*)d_in[1];
  const float* rec_state  = (const float*)d_in[2];
  const float* conv_w     = (const float*)d_in[3];
  const float* qkv_w      = (const float*)d_in[4];
  const float* z_w        = (const float*)d_in[5];
  const float* b_w        = (const float*)d_in[6];
  const float* a_w        = (const float*)d_in[7];
  const float* out_w      = (const float*)d_in[8];
  const float* dt_bias    = (const float*)d_in[9];
  const float* A_log      = (const float*)d_in[10];
  const float* norm_w     = (const float*)d_in[11];

  float* out = (float*)d_out;
  float* out_hidden = out;                         // 2048
  float* out_conv   = out + HDIM;                  // 8192*4
  float* out_rec    = out + HDIM + CONV_DIM * 4;   // 32*128*128

  float* ws = (float*)d_ws;
  float* ws_conv_out = ws;             // 8192
  float* ws_z        = ws + 8192;      // 4096
  float* ws_b        = ws + 12288;     // 32
  float* ws_a        = ws + 12320;     // 32
  float* ws_qn       = ws + 12352;     // 2048
  float* ws_kn       = ws + 14400;     // 2048
  float* ws_beta     = ws + 16448;     // 32
  float* ws_gexp     = ws + 16480;     // 32
  float* ws_xn       = ws + 16512;     // 4096

  hipLaunchKernelGGL(k1_gemv_conv, dim3(ROWS_TOTAL / 8), dim3(256), 0, stream,
                     hidden_in, conv_state, conv_w, qkv_w, z_w, b_w, a_w,
                     out_conv, ws_conv_out, ws_z, ws_b, ws_a);
  hipLaunchKernelGGL(k2_norm_gates, dim3(1), dim3(512), 0, stream, ws_conv_out,
                     ws_b, ws_a, dt_bias, A_log, ws_qn, ws_kn, ws_beta,
                     ws_gexp);
  hipLaunchKernelGGL(k3_rec_wmma, dim3(NV), dim3(256), 0, stream, rec_state,
                     ws_conv_out, ws_z, ws_qn, ws_kn, ws_beta, ws_gexp, norm_w,
                     out_rec, ws_xn);
  hipLaunchKernelGGL(k4_outproj, dim3(HDIM / 8), dim3(256), 0, stream, out_w,
                     ws_xn, out_hidden);
}